// GCN_39573828665589
// MI455X (gfx1250) — compile-verified
//
#include <hip/hip_runtime.h>
#include <hip/hip_bf16.h>

typedef float v2f __attribute__((ext_vector_type(2)));
typedef float v8f __attribute__((ext_vector_type(8)));

#define D_DIM 128

// ---------------------------------------------------------------------------
// Kernel 0: zero the workspace (accumulator [N,128] f32 + indeg [N] u32).
// Harness poisons ws with 0xAA and never re-poisons, so we must clear every call.
// ---------------------------------------------------------------------------
__global__ void gcn_zero_ws(float* __restrict__ ws, int n_dwords) {
    int i = blockIdx.x * blockDim.x + threadIdx.x;
    if (i < n_dwords) ws[i] = 0.0f;
}

// ---------------------------------------------------------------------------
// Kernel 1: edge scatter.  One wave32 per edge: gather h[src] (float4/lane =
// 128 floats), scale by edge weight, atomic-add into acc[dst]; lane 0 bumps
// the in-degree counter.  h and acc are both L2-resident (25.6 MB each).
// ---------------------------------------------------------------------------
__global__ void gcn_edge_scatter(const float* __restrict__ h,
                                 const float* __restrict__ ew,
                                 const int*   __restrict__ src,
                                 const int*   __restrict__ dst,
                                 float*        __restrict__ acc,
                                 unsigned int* __restrict__ indeg,
                                 int n_edges) {
    int wave = (blockIdx.x * blockDim.x + threadIdx.x) >> 5;
    int lane = threadIdx.x & 31;
    if (wave >= n_edges) return;

    int   s = src[wave];
    int   d = dst[wave];
    float w = ew[wave];

    const float4* hv = (const float4*)(h + (size_t)s * D_DIM);
    float4 v = hv[lane];                       // lane covers dims [4*lane, 4*lane+4)

    float* ad = acc + (size_t)d * D_DIM + lane * 4;
    atomicAdd(ad + 0, v.x * w);
    atomicAdd(ad + 1, v.y * w);
    atomicAdd(ad + 2, v.z * w);
    atomicAdd(ad + 3, v.w * w);

    if (lane == 0) atomicAdd(indeg + d, 1u);
}

// ---------------------------------------------------------------------------
// Kernel 2: fused WMMA epilogue.
//   Agg = acc @ Wn ; L = h @ Wl ; E = h @ We   (three 16x16 tiles per wave)
//   out = leaky_relu(Agg * norm + (indeg>0 ? L : E))
// Block = 256 threads = 8 waves; block owns 16 rows, wave w owns cols
// [16w,16w+16).  K=128 swept with V_WMMA_F32_16X16X4_F32 (32 steps).
//
// f32 A-frag layout (16x4): lanes 0-15 hold K = {k0,k0+1} for M=lane,
// lanes 16-31 hold K = {k0+2,k0+3}.  B-frag (4x16) mirrors with N=lane&15.
// D layout: VGPR i -> M = i + 8*(lane>=16), N = lane&15.
// ---------------------------------------------------------------------------
__global__ void __launch_bounds__(256)
gcn_wmma_epilogue(const float* __restrict__ h,
                  const float* __restrict__ acc,
                  const float* __restrict__ norm,
                  const unsigned int* __restrict__ indeg,
                  const float* __restrict__ Wn,
                  const float* __restrict__ Wl,
                  const float* __restrict__ We,
                  float* __restrict__ out,
                  int n_nodes) {
    const int wave = threadIdx.x >> 5;          // 0..7  -> column tile
    const int lane = threadIdx.x & 31;
    const int lm   = lane & 15;
    const int hi   = lane >> 4;                 // 0 or 1

    const int row0 = blockIdx.x * 16;           // 50000/16 = 3125 blocks exactly
    const int col0 = wave * 16;
    const int arow = row0 + lm;                 // A-frag source row for this lane
    const int kh   = 2 * hi;                    // K sub-offset within group of 4

    const float* hrow = h   + (size_t)arow * D_DIM;
    const float* crow = acc + (size_t)arow * D_DIM;

    v8f dAgg = {}; v8f dL = {}; v8f dE = {};

    #pragma unroll 4
    for (int k0 = 0; k0 < D_DIM; k0 += 4) {
        const int ka = k0 + kh;
        v2f aH = *(const v2f*)(hrow + ka);      // h   A-frag (contiguous 8B)
        v2f aC = *(const v2f*)(crow + ka);      // acc A-frag

        const int bc = col0 + lm;
        v2f bN = { Wn[(size_t)ka * D_DIM + bc], Wn[(size_t)(ka + 1) * D_DIM + bc] };
        v2f bL = { Wl[(size_t)ka * D_DIM + bc], Wl[(size_t)(ka + 1) * D_DIM + bc] };
        v2f bE = { We[(size_t)ka * D_DIM + bc], We[(size_t)(ka + 1) * D_DIM + bc] };

        dAgg = __builtin_amdgcn_wmma_f32_16x16x4_f32(false, aC, false, bN,
                                                     (short)0, dAgg, false, false);
        dL   = __builtin_amdgcn_wmma_f32_16x16x4_f32(false, aH, false, bL,
                                                     (short)0, dL, false, false);
        dE   = __builtin_amdgcn_wmma_f32_16x16x4_f32(false, aH, false, bE,
                                                     (short)0, dE, false, false);
    }

    // Epilogue: per-row norm scale, loop/evolve select, leaky ReLU, store.
    #pragma unroll
    for (int i = 0; i < 8; ++i) {
        const int row = row0 + i + 8 * hi;
        const int col = col0 + lm;
        const float loopv = (indeg[row] > 0u) ? dL[i] : dE[i];
        float v = dAgg[i] * norm[row] + loopv;
        v = (v > 0.0f) ? v : 0.01f * v;
        out[(size_t)row * D_DIM + col] = v;
    }
}

// ---------------------------------------------------------------------------
// Launch: inputs are (h, norm, edge_weight, w_neighbor, w_loop, w_evolve,
// src, dst) per setup_inputs() order; all f32 except src/dst (int32).
// ---------------------------------------------------------------------------
extern "C" void kernel_launch(void* const* d_in, const int* in_sizes, int n_in,
                              void* d_out, int out_size, void* d_ws, size_t ws_size,
                              hipStream_t stream) {
    const float* h    = (const float*)d_in[0];
    const float* nrm  = (const float*)d_in[1];
    const float* ew   = (const float*)d_in[2];
    const float* Wn   = (const float*)d_in[3];
    const float* Wl   = (const float*)d_in[4];
    const float* We   = (const float*)d_in[5];
    const int*   src  = (const int*)d_in[6];
    const int*   dst  = (const int*)d_in[7];
    float*       out  = (float*)d_out;

    const int n_nodes = in_sizes[1];            // norm has N elements
    const int n_edges = in_sizes[6];            // src has E elements

    // Workspace layout: acc [N*128] f32, then indeg [N] u32.
    float*        acc   = (float*)d_ws;
    unsigned int* indeg = (unsigned int*)((char*)d_ws + (size_t)n_nodes * D_DIM * sizeof(float));
    const int n_ws_dwords = n_nodes * D_DIM + n_nodes;

    // 1) zero acc + indeg
    {
        int threads = 256;
        int blocks  = (n_ws_dwords + threads - 1) / threads;
        gcn_zero_ws<<<blocks, threads, 0, stream>>>((float*)d_ws, n_ws_dwords);
    }
    // 2) edge gather/scale/scatter-add (one wave per edge)
    {
        int threads = 256;                      // 8 waves -> 8 edges per block
        int blocks  = (n_edges + 7) / 8;
        gcn_edge_scatter<<<blocks, threads, 0, stream>>>(h, ew, src, dst,
                                                         acc, indeg, n_edges);
    }
    // 3) fused triple-GEMM + epilogue (16 rows per block, 8 col-tile waves)
    {
        int threads = 256;
        int blocks  = (n_nodes + 15) / 16;      // 3125
        gcn_wmma_epilogue<<<blocks, threads, 0, stream>>>(h, acc, nrm, indeg,
                                                          Wn, Wl, We, out, n_nodes);
    }
}